// PartialEncoderWeightedSumEDDIMultiWeightFast_57767310131613
// MI455X (gfx1250) — compile-verified
//
#include <hip/hip_runtime.h>
#include <math.h>

// ---------------------------------------------------------------------------
// PartialEncoderWeightedSumEDDIMultiWeightFast for MI455X (gfx1250, wave32).
//
// Roofline: ~46 GFLOP fp32 total, ~23 MB of unique input traffic (HBM at
// 23.3 TB/s -> ~1us of mandatory traffic). Compute-dominated but tiny.
// Strategy: single fused pass over J with online softmax (no 327 MB h_out
// round-trip), f32 WMMA 16x16x4 for the two dominant GEMMs, partial
// {max,sum,acc} records per (b, j-chunk, head) combined by a second kernel
// that also runs the tiny per-batch tail MLPs.
// ---------------------------------------------------------------------------

typedef __attribute__((ext_vector_type(2))) float v2f;
typedef __attribute__((ext_vector_type(8))) float v8f;

#define B_N     128
#define J_N     20000
#define D_N     32
#define W_N     4
#define HH_N    128
#define CHUNK_J 128              // j's per block (2 waves x 4 tiles x 16 rows)
#define NCHUNK  157              // ceil(20000/128)
#define PREC    34               // floats per partial record {m, s, a[32]}
#define NEGBIG  (-1.0e9f)
#define NEGINF  (-3.0e38f)
#define EPS_LN  1e-5f

__device__ __forceinline__ v8f wmma4(v2f a, v2f b, v8f c) {
  // D = A(16x4,f32) * B(4x16,f32) + C(16x16,f32)  ->  v_wmma_f32_16x16x4_f32
  return __builtin_amdgcn_wmma_f32_16x16x4_f32(false, a, false, b, (short)0, c,
                                               false, false);
}

// ---------------------------------------------------------------------------
// Kernel 1: fused h-layer MLP + gates + online softmax partials.
// grid = (NCHUNK, B), block = 64 threads (2 waves). Each wave processes four
// 16-row j tiles with WMMA; block writes one partial record per head to ws.
// ---------------------------------------------------------------------------
__global__ __launch_bounds__(64) void k_main(
    const float* __restrict__ x, const int* __restrict__ mask,
    const float* __restrict__ Femb,
    const float* __restrict__ w1, const float* __restrict__ b1,
    const float* __restrict__ ln1g, const float* __restrict__ ln1b,
    const float* __restrict__ w2, const float* __restrict__ b2,
    const float* __restrict__ ln2g, const float* __restrict__ ln2b,
    const float* __restrict__ gw1, const float* __restrict__ gb1,
    const float* __restrict__ gw2, const float* __restrict__ gb2,
    float* __restrict__ part)
{
  __shared__ float sW1[33 * 128];
  __shared__ float sB1[128], sG1[128], sBt1[128];
  __shared__ float sW2[128 * 32];
  __shared__ float sB2[32], sG2[32], sBt2[32];
  __shared__ float sGW1[32 * 16], sGB1[16], sGW2[16 * 4], sGB2[4];
  __shared__ float sH1[2][16 * 132];   // per-wave h1 staging (pad 132 vs banks)
  __shared__ float sHO[2][16 * 33];    // per-wave h_out staging
  __shared__ float sPart[2][W_N][PREC];

  const int tid   = threadIdx.x;
  const int wave  = tid >> 5;
  const int lane  = tid & 31;
  const int m16   = lane & 15;
  const int khalf = lane >> 4;          // which K/row half of the wave
  const int chunk = blockIdx.x;
  const int b     = blockIdx.y;

  // ---- stage weights in LDS (reused by all 8 j-tiles of this block) ----
  for (int i = tid; i < 33 * 128; i += 64) sW1[i] = w1[i];
  for (int i = tid; i < 128; i += 64) { sB1[i] = b1[i]; sG1[i] = ln1g[i]; sBt1[i] = ln1b[i]; }
  for (int i = tid; i < 128 * 32; i += 64) sW2[i] = w2[i];
  if (tid < 32) { sB2[tid] = b2[tid]; sG2[tid] = ln2g[tid]; sBt2[tid] = ln2b[tid]; }
  for (int i = tid; i < 512; i += 64) sGW1[i] = gw1[i];
  if (tid < 16) sGB1[tid] = gb1[tid];
  if (tid < 64) sGW2[tid] = gw2[tid];
  if (tid < 4)  sGB2[tid] = gb2[tid];
  __syncthreads();

  float m_run[W_N], s_run[W_N], acc[W_N][2];
  #pragma unroll
  for (int w = 0; w < W_N; ++w) {
    m_run[w] = NEGINF; s_run[w] = 0.f; acc[w][0] = 0.f; acc[w][1] = 0.f;
  }

  float* sH1w = sH1[wave];
  float* sHOw = sHO[wave];

  for (int t = 0; t < 4; ++t) {
    const int j0 = chunk * CHUNK_J + wave * 64 + t * 16;
    const int jr = j0 + m16;
    const bool vj = (jr < J_N);
    const int jc = vj ? jr : (J_N - 1);

    // ---- A-matrix fragments: F_emb rows (16x32 f32, K chunks of 4) ----
    // A layout: lanes 0-15 hold K={4kc,4kc+1}, lanes 16-31 K={4kc+2,4kc+3}.
    v2f afr[8];
    const float* fp = Femb + (size_t)jc * 32 + khalf * 2;
    #pragma unroll
    for (int kc = 0; kc < 8; ++kc) afr[kc] = *(const v2f*)(fp + kc * 4);

    const float xv  = x[(size_t)b * J_N + jc];
    const float mvf = (vj && mask[(size_t)b * J_N + jc] != 0) ? 1.f : 0.f;

    // ---- matmul 1: out1[16x128] = [x|F] @ w1[33x128] + b1 ----
    v8f o1[8];
    #pragma unroll
    for (int nt = 0; nt < 8; ++nt) {
      v8f c = {0.f, 0.f, 0.f, 0.f, 0.f, 0.f, 0.f, 0.f};
      const int col = nt * 16 + m16;
      #pragma unroll
      for (int kc = 0; kc < 8; ++kc) {
        const int krow = 1 + kc * 4 + khalf * 2;   // w1 rows 1..32 = F_emb dims
        v2f bf;
        bf.x = sW1[krow * 128 + col];
        bf.y = sW1[(krow + 1) * 128 + col];
        c = wmma4(afr[kc], bf, c);
      }
      const float w0  = sW1[col];     // w1 row 0 multiplies x (rank-1 update)
      const float bia = sB1[col];
      #pragma unroll
      for (int r = 0; r < 8; ++r) {
        const float xr = __shfl(xv, r + (khalf << 3), 32); // x[row(r,half)]
        c[r] += xr * w0 + bia;
      }
      o1[nt] = c;
    }

    // ---- LN1 over 128 cols per row + ReLU, staged to LDS ----
    #pragma unroll
    for (int r = 0; r < 8; ++r) {
      float s = 0.f, q = 0.f;
      #pragma unroll
      for (int nt = 0; nt < 8; ++nt) { const float v = o1[nt][r]; s += v; q += v * v; }
      #pragma unroll
      for (int mk = 1; mk < 16; mk <<= 1) {
        s += __shfl_xor(s, mk, 32); q += __shfl_xor(q, mk, 32);
      }
      const float mean = s * (1.f / 128.f);
      const float var  = q * (1.f / 128.f) - mean * mean;
      const float rstd = rsqrtf(var + EPS_LN);
      const int row = r + (khalf << 3);
      #pragma unroll
      for (int nt = 0; nt < 8; ++nt) {
        const int col = nt * 16 + m16;
        float v = (o1[nt][r] - mean) * rstd * sG1[col] + sBt1[col];
        sH1w[row * 132 + col] = fmaxf(v, 0.f);
      }
    }
    // wave-private staging: same-wave LDS ops are in-order, no barrier needed.

    // ---- matmul 2: out2[16x32] = h1[16x128] @ w2[128x32] + b2 ----
    v8f o2[2];
    #pragma unroll
    for (int nt2 = 0; nt2 < 2; ++nt2) {
      v8f c = {0.f, 0.f, 0.f, 0.f, 0.f, 0.f, 0.f, 0.f};
      const int col = nt2 * 16 + m16;
      #pragma unroll
      for (int kc = 0; kc < 32; ++kc) {
        v2f a2 = *(const v2f*)(&sH1w[m16 * 132 + kc * 4 + khalf * 2]);
        const int krow = kc * 4 + khalf * 2;
        v2f bf;
        bf.x = sW2[krow * 32 + col];
        bf.y = sW2[(krow + 1) * 32 + col];
        c = wmma4(a2, bf, c);
      }
      const float bia = sB2[col];
      #pragma unroll
      for (int r = 0; r < 8; ++r) c[r] += bia;
      o2[nt2] = c;
    }

    // ---- LN2 over 32 cols per row + ReLU; keep frags + stage rows ----
    #pragma unroll
    for (int r = 0; r < 8; ++r) {
      float s = 0.f, q = 0.f;
      #pragma unroll
      for (int nt2 = 0; nt2 < 2; ++nt2) { const float v = o2[nt2][r]; s += v; q += v * v; }
      #pragma unroll
      for (int mk = 1; mk < 16; mk <<= 1) {
        s += __shfl_xor(s, mk, 32); q += __shfl_xor(q, mk, 32);
      }
      const float mean = s * (1.f / 32.f);
      const float var  = q * (1.f / 32.f) - mean * mean;
      const float rstd = rsqrtf(var + EPS_LN);
      const int row = r + (khalf << 3);
      #pragma unroll
      for (int nt2 = 0; nt2 < 2; ++nt2) {
        const int col = nt2 * 16 + m16;
        float v = (o2[nt2][r] - mean) * rstd * sG2[col] + sBt2[col];
        v = fmaxf(v, 0.f);
        o2[nt2][r] = v;                 // h_out (unmasked: gates use this)
        sHOw[row * 33 + col] = v;
      }
    }

    // ---- gate head per row (VALU; 576 FLOP/row, lanes mirror per half) ----
    float g1v[16];
    const float* hrow = &sHOw[m16 * 33];
    #pragma unroll
    for (int o = 0; o < 16; ++o) {
      float tacc = sGB1[o];
      #pragma unroll
      for (int k = 0; k < 32; ++k) tacc += hrow[k] * sGW1[k * 16 + o];
      g1v[o] = fmaxf(tacc, 0.f);
    }
    float lg[W_N];
    #pragma unroll
    for (int w = 0; w < W_N; ++w) {
      float tacc = sGB2[w];
      #pragma unroll
      for (int k = 0; k < 16; ++k) tacc += g1v[k] * sGW2[k * 4 + w];
      lg[w] = (mvf > 0.f) ? tacc : NEGBIG;   // mask + out-of-range j
    }

    // ---- online softmax update per head ----
    #pragma unroll
    for (int w = 0; w < W_N; ++w) {
      float tl = (lane < 16) ? lg[w] : NEGINF;
      #pragma unroll
      for (int mk = 1; mk < 32; mk <<= 1) tl = fmaxf(tl, __shfl_xor(tl, mk, 32));
      const float mnew  = fmaxf(m_run[w], tl);
      const float scale = __expf(m_run[w] - mnew);
      float p = (lane < 16) ? __expf(lg[w] - mnew) : 0.f;
      float ps = p;
      #pragma unroll
      for (int mk = 1; mk < 32; mk <<= 1) ps += __shfl_xor(ps, mk, 32);
      s_run[w] = s_run[w] * scale + ps;
      m_run[w] = mnew;
      const float pm = p * mvf;            // h_masked: zero masked rows
      #pragma unroll
      for (int nt2 = 0; nt2 < 2; ++nt2) {
        float a = acc[w][nt2] * scale;
        #pragma unroll
        for (int r = 0; r < 8; ++r) {
          const float pr = __shfl(pm, r + (khalf << 3), 32); // p[row(r,half)]
          a = fmaf(pr, o2[nt2][r], a);
        }
        acc[w][nt2] = a;
      }
    }
  }

  // ---- merge the two row-halves of the accumulators ----
  #pragma unroll
  for (int w = 0; w < W_N; ++w) {
    acc[w][0] += __shfl_xor(acc[w][0], 16, 32);
    acc[w][1] += __shfl_xor(acc[w][1], 16, 32);
    if (lane == 0) { sPart[wave][w][0] = m_run[w]; sPart[wave][w][1] = s_run[w]; }
    if (lane < 16) {
      sPart[wave][w][2 + lane]  = acc[w][0];   // d = lane
      sPart[wave][w][18 + lane] = acc[w][1];   // d = 16 + lane
    }
  }
  __syncthreads();

  // ---- combine 2 wave partials; each wave finalizes 2 heads ----
  #pragma unroll
  for (int hh = 0; hh < 2; ++hh) {
    const int w = wave * 2 + hh;
    const float m0 = sPart[0][w][0], m1 = sPart[1][w][0];
    const float M  = fmaxf(m0, m1);
    const float e0 = __expf(m0 - M), e1 = __expf(m1 - M);
    float* out = part + (((size_t)b * NCHUNK + chunk) * W_N + w) * PREC;
    if (lane == 0) {
      out[0] = M;
      out[1] = sPart[0][w][1] * e0 + sPart[1][w][1] * e1;
    }
    out[2 + lane] = sPart[0][w][2 + lane] * e0 + sPart[1][w][2 + lane] * e1;
  }
}

// ---------------------------------------------------------------------------
// Kernel 2: combine chunk partials -> head_sums, then combiner + encoder
// MLPs. One wave per batch row; everything via lane shuffles (4 MFLOP total).
// ---------------------------------------------------------------------------
__global__ __launch_bounds__(32) void k_tail(
    const float* __restrict__ part,
    const float* __restrict__ cw, const float* __restrict__ cb,
    const float* __restrict__ clng, const float* __restrict__ clnb,
    const float* __restrict__ ew1, const float* __restrict__ eb1,
    const float* __restrict__ e1g, const float* __restrict__ e1b,
    const float* __restrict__ ew2, const float* __restrict__ eb2,
    const float* __restrict__ e2g, const float* __restrict__ e2b,
    float* __restrict__ out)
{
  const int b = blockIdx.x;
  const int lane = threadIdx.x;

  // ---- global softmax combine: head_sums[w][d], lane holds d = lane ----
  float hs[W_N];
  #pragma unroll
  for (int w = 0; w < W_N; ++w) {
    const float* pb = part + ((size_t)b * NCHUNK * W_N + w) * PREC;
    float M = NEGINF;
    for (int c = 0; c < NCHUNK; ++c) M = fmaxf(M, pb[(size_t)c * W_N * PREC]);
    float S = 0.f, A = 0.f;
    for (int c = 0; c < NCHUNK; ++c) {
      const float* r = pb + (size_t)c * W_N * PREC;
      const float e = __expf(r[0] - M);
      S += r[1] * e;
      A += r[2 + lane] * e;
    }
    hs[w] = A / S;
  }

  // ---- combiner: relu(LN(hs_flat[128] @ cw[128x32] + cb)) ----
  float comb;
  {
    float tacc = cb[lane];
    #pragma unroll
    for (int w = 0; w < W_N; ++w)
      for (int d = 0; d < 32; ++d)
        tacc += __shfl(hs[w], d, 32) * cw[(w * 32 + d) * 32 + lane];
    float s = tacc, q = tacc * tacc;
    #pragma unroll
    for (int mk = 1; mk < 32; mk <<= 1) { s += __shfl_xor(s, mk, 32); q += __shfl_xor(q, mk, 32); }
    const float mean = s * (1.f / 32.f);
    const float rstd = rsqrtf(q * (1.f / 32.f) - mean * mean + EPS_LN);
    comb = fmaxf((tacc - mean) * rstd * clng[lane] + clnb[lane], 0.f);
  }

  // ---- encoder layer 1: relu(LN(comb[32] @ ew1[32x128] + eb1)) ----
  float e1v[4];
  {
    float tacc[4];
    #pragma unroll
    for (int i = 0; i < 4; ++i) tacc[i] = eb1[lane + 32 * i];
    for (int k = 0; k < 32; ++k) {
      const float ck = __shfl(comb, k, 32);
      #pragma unroll
      for (int i = 0; i < 4; ++i) tacc[i] += ck * ew1[k * 128 + lane + 32 * i];
    }
    float s = 0.f, q = 0.f;
    #pragma unroll
    for (int i = 0; i < 4; ++i) { s += tacc[i]; q += tacc[i] * tacc[i]; }
    #pragma unroll
    for (int mk = 1; mk < 32; mk <<= 1) { s += __shfl_xor(s, mk, 32); q += __shfl_xor(q, mk, 32); }
    const float mean = s * (1.f / 128.f);
    const float rstd = rsqrtf(q * (1.f / 128.f) - mean * mean + EPS_LN);
    #pragma unroll
    for (int i = 0; i < 4; ++i) {
      const int o = lane + 32 * i;
      e1v[i] = fmaxf((tacc[i] - mean) * rstd * e1g[o] + e1b[o], 0.f);
    }
  }

  // ---- encoder layer 2: relu(LN(e1[128] @ ew2[128x64] + eb2)) -> mu|logvar
  {
    float t0 = eb2[lane], t1 = eb2[lane + 32];
    #pragma unroll
    for (int i4 = 0; i4 < 4; ++i4)
      for (int kk = 0; kk < 32; ++kk) {
        const float ek = __shfl(e1v[i4], kk, 32);
        const int k = i4 * 32 + kk;
        t0 += ek * ew2[k * 64 + lane];
        t1 += ek * ew2[k * 64 + lane + 32];
      }
    float s = t0 + t1, q = t0 * t0 + t1 * t1;
    #pragma unroll
    for (int mk = 1; mk < 32; mk <<= 1) { s += __shfl_xor(s, mk, 32); q += __shfl_xor(q, mk, 32); }
    const float mean = s * (1.f / 64.f);
    const float rstd = rsqrtf(q * (1.f / 64.f) - mean * mean + EPS_LN);
    const float mu = fmaxf((t0 - mean) * rstd * e2g[lane] + e2b[lane], 0.f);
    const float lv = fmaxf((t1 - mean) * rstd * e2g[lane + 32] + e2b[lane + 32], 0.f);
    out[b * 32 + lane] = mu;                    // mu  [B,32]
    out[B_N * 32 + b * 32 + lane] = lv;         // logvar [B,32]
  }
}

extern "C" void kernel_launch(void* const* d_in, const int* in_sizes, int n_in,
                              void* d_out, int out_size, void* d_ws, size_t ws_size,
                              hipStream_t stream) {
  (void)in_sizes; (void)n_in; (void)out_size; (void)ws_size;
  const float* x     = (const float*)d_in[0];
  const int*   mask  = (const int*)  d_in[1];
  const float* Femb  = (const float*)d_in[2];
  const float* w1    = (const float*)d_in[3];
  const float* b1    = (const float*)d_in[4];
  const float* ln1g  = (const float*)d_in[5];
  const float* ln1b  = (const float*)d_in[6];
  const float* w2    = (const float*)d_in[7];
  const float* b2    = (const float*)d_in[8];
  const float* ln2g  = (const float*)d_in[9];
  const float* ln2b  = (const float*)d_in[10];
  const float* gw1   = (const float*)d_in[11];
  const float* gb1   = (const float*)d_in[12];
  const float* gw2   = (const float*)d_in[13];
  const float* gb2   = (const float*)d_in[14];
  const float* cw    = (const float*)d_in[15];
  const float* cb    = (const float*)d_in[16];
  const float* clng  = (const float*)d_in[17];
  const float* clnb  = (const float*)d_in[18];
  const float* ew1   = (const float*)d_in[19];
  const float* eb1   = (const float*)d_in[20];
  const float* e1g   = (const float*)d_in[21];
  const float* e1b   = (const float*)d_in[22];
  const float* ew2   = (const float*)d_in[23];
  const float* eb2   = (const float*)d_in[24];
  const float* e2g   = (const float*)d_in[25];
  const float* e2b   = (const float*)d_in[26];

  float* part = (float*)d_ws;   // B*NCHUNK*W*PREC floats ~= 10.4 MB

  dim3 g1(NCHUNK, B_N);
  k_main<<<g1, 64, 0, stream>>>(x, mask, Femb, w1, b1, ln1g, ln1b,
                                w2, b2, ln2g, ln2b, gw1, gb1, gw2, gb2, part);
  k_tail<<<B_N, 32, 0, stream>>>(part, cw, cb, clng, clnb,
                                 ew1, eb1, e1g, e1b, ew2, eb2, e2g, e2b,
                                 (float*)d_out);
}